// ExposureAwareAttention_11914239279387
// MI455X (gfx1250) — compile-verified
//
#include <hip/hip_runtime.h>

typedef __attribute__((ext_vector_type(2))) float v2f;
typedef __attribute__((ext_vector_type(8))) float v8f;

#define HW      (512*512)     // elements per (b,c) plane
#define PLANES  512           // B*C = 8*64
#define NSEG    8             // segments per plane
#define SEGF    (HW/NSEG)     // 32768 floats per segment
#define SEGV    (SEGF/4)      // 8192 float4 per segment

// ---------------------------------------------------------------------------
// Kernel 1: per-(plane, segment) partial sum + max.  Bandwidth-bound pass #1.
// ---------------------------------------------------------------------------
__global__ __launch_bounds__(256) void k_reduce(const float* __restrict__ x,
                                                float* __restrict__ psum,
                                                float* __restrict__ pmax) {
  const int plane = blockIdx.x >> 3;
  const int seg   = blockIdx.x & 7;
  const float4* xv = (const float4*)(x + (size_t)plane * HW + (size_t)seg * SEGF);
  float s = 0.f;
  float m = -__builtin_inff();
#pragma unroll 4
  for (int i = 0; i < SEGV / 256; ++i) {
    float4 v = xv[i * 256 + threadIdx.x];
    s += v.x + v.y + v.z + v.w;
    m = fmaxf(m, fmaxf(fmaxf(v.x, v.y), fmaxf(v.z, v.w)));
  }
  __shared__ float ls[256];
  __shared__ float lm[256];
  ls[threadIdx.x] = s;
  lm[threadIdx.x] = m;
  __syncthreads();
  for (int off = 128; off > 0; off >>= 1) {
    if (threadIdx.x < (unsigned)off) {
      ls[threadIdx.x] += ls[threadIdx.x + off];
      lm[threadIdx.x] = fmaxf(lm[threadIdx.x], lm[threadIdx.x + off]);
    }
    __syncthreads();
  }
  if (threadIdx.x == 0) {
    psum[blockIdx.x] = ls[0];
    pmax[blockIdx.x] = lm[0];
  }
}

// ---------------------------------------------------------------------------
// Kernel 2: one wave32.  Finish reduction, then both 1x1-conv matmuls with
// V_WMMA_F32_16X16X4_F32 (fp32 WMMA keeps reference precision).
//   Stage A: H[16,16] = combined[16,128] x w1^T[128,16]   (32 WMMAs, K=4 each)
//   Stage B: G[16,64] = H[16,8] x w2^T[8,64]              (4 tiles x 2 WMMAs)
// A-operand layout (ISA 7.12.2, 32-bit 16x4): lanes 0-15 -> K={0,1},
// lanes 16-31 -> K={2,3}; B/C/D: one row striped across lanes per VGPR.
// ---------------------------------------------------------------------------
__global__ __launch_bounds__(32) void k_gate(const float* __restrict__ psum,
                                             const float* __restrict__ pmax,
                                             const float* __restrict__ w1,
                                             const float* __restrict__ w2,
                                             float* __restrict__ gate) {
  __shared__ float comb[16][128];   // [batch padded to 16][2C]
  __shared__ float Hmat[16][16];    // leaky-relu'd hidden, padded

  const int lane = threadIdx.x;
  for (int i = lane; i < 16 * 128; i += 32) ((float*)comb)[i] = 0.f;
  __syncthreads();

  // Finish reduction: 16 planes per lane (512 planes / 32 lanes).
  for (int j = 0; j < 16; ++j) {
    const int p = lane * 16 + j;
    float s = 0.f;
    float m = -__builtin_inff();
#pragma unroll
    for (int g = 0; g < NSEG; ++g) {
      s += psum[p * NSEG + g];
      m = fmaxf(m, pmax[p * NSEG + g]);
    }
    const int b = p >> 6, c = p & 63;
    comb[b][c]      = s * (1.f / (float)HW);  // mean
    comb[b][64 + c] = m;                      // max
  }
  __syncthreads();

  const int half = lane >> 4;    // selects K pair within the 16x4 A tile
  const int mrow = lane & 15;    // M for A-operand, N for B/C/D operands

  // ---- Stage A: combined @ w1^T, K = 128 in 32 steps of 4 ----
  v8f accH = {};
#pragma unroll
  for (int k = 0; k < 32; ++k) {
    const int kb = 4 * k + 2 * half;
    v2f a, b;
    a.x = comb[mrow][kb];
    a.y = comb[mrow][kb + 1];
    float b0 = 0.f, b1 = 0.f;
    if (mrow < 8) {                 // N padding: w1 has only 8 rows
      b0 = w1[mrow * 128 + kb];     // B[K][N] = w1[N][K]
      b1 = w1[mrow * 128 + kb + 1];
    }
    b.x = b0; b.y = b1;
    accH = __builtin_amdgcn_wmma_f32_16x16x4_f32(
        false, a, false, b, (short)0, accH, false, false);
  }

  // LeakyReLU(0.2) + relayout D (lane-striped rows) -> LDS for next A-operand.
#pragma unroll
  for (int v = 0; v < 8; ++v) {
    float h = accH[v];
    h = (h >= 0.f) ? h : 0.2f * h;
    Hmat[v + 8 * half][mrow] = h;   // D: VGPR v = row M (=v or v+8), lane = col N
  }
  __syncthreads();

  // ---- Stage B: H @ w2^T, N = 64 as four 16-wide tiles, K = 8 in 2 steps ----
#pragma unroll
  for (int nt = 0; nt < 4; ++nt) {
    v8f accG = {};
#pragma unroll
    for (int kc = 0; kc < 2; ++kc) {
      const int kb = 4 * kc + 2 * half;
      v2f a, b;
      a.x = Hmat[mrow][kb];
      a.y = Hmat[mrow][kb + 1];
      const int n = nt * 16 + mrow;   // global output channel, always < 64
      b.x = w2[n * 8 + kb];           // B[K][N] = w2[N][K]
      b.y = w2[n * 8 + kb + 1];
      accG = __builtin_amdgcn_wmma_f32_16x16x4_f32(
          false, a, false, b, (short)0, accG, false, false);
    }
    // D rows 0..7 (batches) live in VGPRs 0..7 of lanes 0-15.
    if (half == 0) {
#pragma unroll
      for (int v = 0; v < 8; ++v) {
        const float att = 1.f / (1.f + __expf(-accG[v]));
        gate[v * 64 + nt * 16 + mrow] = att * 0.4f + 0.8f;
      }
    }
  }
}

// ---------------------------------------------------------------------------
// Kernel 3: out = x * gate[plane].  Bandwidth-bound pass #2 (read + write).
// gate[plane] is blockIdx-uniform -> scalar load.
// ---------------------------------------------------------------------------
__global__ __launch_bounds__(256) void k_scale(const float* __restrict__ x,
                                               const float* __restrict__ gate,
                                               float* __restrict__ out) {
  const int plane = blockIdx.x >> 3;
  const int seg   = blockIdx.x & 7;
  const float g = gate[plane];
  const size_t base = (size_t)plane * HW + (size_t)seg * SEGF;
  const float4* xv = (const float4*)(x + base);
  float4* ov = (float4*)(out + base);
#pragma unroll 4
  for (int i = 0; i < SEGV / 256; ++i) {
    float4 v = xv[i * 256 + threadIdx.x];
    v.x *= g; v.y *= g; v.z *= g; v.w *= g;
    ov[i * 256 + threadIdx.x] = v;
  }
}

extern "C" void kernel_launch(void* const* d_in, const int* in_sizes, int n_in,
                              void* d_out, int out_size, void* d_ws, size_t ws_size,
                              hipStream_t stream) {
  const float* x  = (const float*)d_in[0];   // [8,64,512,512]
  const float* w1 = (const float*)d_in[1];   // [8,128]
  const float* w2 = (const float*)d_in[2];   // [64,8]
  float* out = (float*)d_out;

  float* psum = (float*)d_ws;                // PLANES*NSEG floats
  float* pmax = psum + PLANES * NSEG;        // PLANES*NSEG floats
  float* gate = pmax + PLANES * NSEG;        // 512 floats

  k_reduce<<<PLANES * NSEG, 256, 0, stream>>>(x, psum, pmax);
  k_gate<<<1, 32, 0, stream>>>(psum, pmax, w1, w2, gate);
  k_scale<<<PLANES * NSEG, 256, 0, stream>>>(x, gate, out);
}